// CPPN_70660801954511
// MI455X (gfx1250) — compile-verified
//
#include <hip/hip_runtime.h>

// CPPN MLP: [P,12] -> 32 -> 32 -> 32 -> 3 on 1M pixels.
// Memory floor ~60MB @ 23.3TB/s = ~2.6us; GEMMs via V_WMMA_F32_16X16X4_F32
// (exact f32 semantics); activation VALU work is the real bottleneck, so the
// five CPPN activations are folded into one exp2/rcp form selected by
// per-lane constants (activation id is uniform per lane in WMMA C-layout).

typedef __attribute__((ext_vector_type(2))) float v2f;
typedef __attribute__((ext_vector_type(8))) float v8f;

#define P_PIX   (1024 * 1024)
#define NTILES  (P_PIX / 16)
#define LSTR    34   // LDS row stride in floats: even (b64-aligned) & conflict-free

struct Coef { float A, B, S, F, C, bias; int id; };

__device__ __forceinline__ Coef make_coef(int id, float bias) {
  const float L2E = 1.4426950408889634f;
  Coef c; c.A = 0.f; c.B = 0.f; c.S = 0.f; c.F = 1.f; c.C = 0.f; c.id = id; c.bias = bias;
  if (id == 1)      { c.A =  2.f * L2E; c.S = -2.f; c.C = 1.f; }  // tanh = 1 - 2/(e^2x + 1)
  else if (id == 2) { c.A = -L2E;       c.S =  1.f; }             // sigmoid = 1/(e^-x + 1)
  else if (id == 4) { c.B =  0.5f*L2E;  c.S =  1.f; c.F = 0.f; }  // gauss = 1/e^(x^2/2)
  return c;
}

__device__ __forceinline__ float apply_act(float x, const Coef& c) {
  float t  = __builtin_amdgcn_exp2f(__builtin_fmaf(c.B, x * x, c.A * x));
  float y  = __builtin_fmaf(c.S, __builtin_amdgcn_rcpf(t + c.F), c.C);
  float sn = __builtin_amdgcn_sinf(x * 0.15915494309189535f); // v_sin takes x/(2*pi)
  y = (c.id == 3) ? sn : y;
  y = (c.id == 0) ? x  : y;
  return y;
}

__global__ __launch_bounds__(256) void cppn_kernel(
    const float* __restrict__ inputs, const float* __restrict__ bias_in,
    const float* __restrict__ W1, const float* __restrict__ b1, const int* __restrict__ act1,
    const float* __restrict__ W2, const float* __restrict__ b2, const int* __restrict__ act2,
    const float* __restrict__ W3, const float* __restrict__ b3, const int* __restrict__ act3,
    const float* __restrict__ Wout, const float* __restrict__ bout,
    float* __restrict__ out)
{
  __shared__ float lds[8][16 * LSTR];           // 17.4 KB, wave-private strips
  const int lane = threadIdx.x & 31;
  const int w    = threadIdx.x >> 5;
  const int n0   = lane & 15;                   // column (B/C/D) or pixel row (A)
  const int kh   = lane >> 4;                   // K-half selector in A/B frags
  float* myLds   = &lds[w][0];

  // Per-lane activation coefficients + biases (column n = n0 + 16*t).
  Coef c1[2], c2[2], c3[2];
#pragma unroll
  for (int t = 0; t < 2; ++t) {
    int n = n0 + 16 * t;
    c1[t] = make_coef(act1[n], b1[n]);
    c2[t] = make_coef(act2[n], b2[n]);
    c3[t] = make_coef(act3[n], b3[n]);
  }
  const Coef cOut = make_coef(1, (n0 < 3) ? bout[n0] : 0.f); // output = tanh

  // Input-node bias pairs for A-fragments (K = 4*kb + 2*kh + {0,1}).
  v2f bin[3];
#pragma unroll
  for (int kb = 0; kb < 3; ++kb) {
    int k = 4 * kb + 2 * kh;
    bin[kb][0] = bias_in[k]; bin[kb][1] = bias_in[k + 1];
  }

  // Register-resident WMMA B-fragments for all weights.
  // B frag (vgpr r, lane l): W[4*kb + 2*(l>>4) + r][ (l&15) + 16*t ]
  v2f fW1[3][2], fW2[8][2], fW3[8][2], fWo[8];
#pragma unroll
  for (int kb = 0; kb < 3; ++kb) {
    int k = 4 * kb + 2 * kh;
#pragma unroll
    for (int t = 0; t < 2; ++t) {
      int n = n0 + 16 * t;
      fW1[kb][t][0] = W1[k * 32 + n]; fW1[kb][t][1] = W1[(k + 1) * 32 + n];
    }
  }
#pragma unroll
  for (int kb = 0; kb < 8; ++kb) {
    int k = 4 * kb + 2 * kh;
#pragma unroll
    for (int t = 0; t < 2; ++t) {
      int n = n0 + 16 * t;
      fW2[kb][t][0] = W2[k * 32 + n]; fW2[kb][t][1] = W2[(k + 1) * 32 + n];
      fW3[kb][t][0] = W3[k * 32 + n]; fW3[kb][t][1] = W3[(k + 1) * 32 + n];
    }
    // Wout is [32,3]; pad N to 16 with zeros.
    fWo[kb][0] = (n0 < 3) ? Wout[k * 3 + n0]       : 0.f;
    fWo[kb][1] = (n0 < 3) ? Wout[(k + 1) * 3 + n0] : 0.f;
  }

  const int gw = blockIdx.x * 8 + w;
  const int nw = gridDim.x * 8;

  for (int tile = gw; tile < NTILES; tile += nw) {
    const int p0 = tile * 16;

    // ---- A-fragments of the input tile (16 pixels x K=12), + input bias ----
    v2f aX[3];
    const float* row = inputs + (size_t)(p0 + n0) * 12;
#pragma unroll
    for (int kb = 0; kb < 3; ++kb) {
      int k = 4 * kb + 2 * kh;
      v2f a = *(const v2f*)(row + k);
      aX[kb] = a + bin[kb];
    }

    // ---- layer 1: h1 = act1(X @ W1 + b1) ----
#pragma unroll
    for (int t = 0; t < 2; ++t) {
      v8f acc = {0.f, 0.f, 0.f, 0.f, 0.f, 0.f, 0.f, 0.f};
#pragma unroll
      for (int kb = 0; kb < 3; ++kb)
        acc = __builtin_amdgcn_wmma_f32_16x16x4_f32(
            false, aX[kb], false, fW1[kb][t], (short)0, acc, false, false);
#pragma unroll
      for (int i = 0; i < 8; ++i) {
        float v = apply_act(acc[i] + c1[t].bias, c1[t]);
        myLds[(i + 8 * kh) * LSTR + n0 + 16 * t] = v;   // C-layout -> LDS
      }
    }
    __builtin_amdgcn_wave_barrier();

    // ---- reload h1 in A-layout ----
    v2f aH[8];
#pragma unroll
    for (int kb = 0; kb < 8; ++kb)
      aH[kb] = *(const v2f*)&myLds[n0 * LSTR + 4 * kb + 2 * kh];

    // ---- layer 2 ----
#pragma unroll
    for (int t = 0; t < 2; ++t) {
      v8f acc = {0.f, 0.f, 0.f, 0.f, 0.f, 0.f, 0.f, 0.f};
#pragma unroll
      for (int kb = 0; kb < 8; ++kb)
        acc = __builtin_amdgcn_wmma_f32_16x16x4_f32(
            false, aH[kb], false, fW2[kb][t], (short)0, acc, false, false);
#pragma unroll
      for (int i = 0; i < 8; ++i) {
        float v = apply_act(acc[i] + c2[t].bias, c2[t]);
        myLds[(i + 8 * kh) * LSTR + n0 + 16 * t] = v;
      }
    }
    __builtin_amdgcn_wave_barrier();
#pragma unroll
    for (int kb = 0; kb < 8; ++kb)
      aH[kb] = *(const v2f*)&myLds[n0 * LSTR + 4 * kb + 2 * kh];

    // ---- layer 3 ----
#pragma unroll
    for (int t = 0; t < 2; ++t) {
      v8f acc = {0.f, 0.f, 0.f, 0.f, 0.f, 0.f, 0.f, 0.f};
#pragma unroll
      for (int kb = 0; kb < 8; ++kb)
        acc = __builtin_amdgcn_wmma_f32_16x16x4_f32(
            false, aH[kb], false, fW3[kb][t], (short)0, acc, false, false);
#pragma unroll
      for (int i = 0; i < 8; ++i) {
        float v = apply_act(acc[i] + c3[t].bias, c3[t]);
        myLds[(i + 8 * kh) * LSTR + n0 + 16 * t] = v;
      }
    }
    __builtin_amdgcn_wave_barrier();
#pragma unroll
    for (int kb = 0; kb < 8; ++kb)
      aH[kb] = *(const v2f*)&myLds[n0 * LSTR + 4 * kb + 2 * kh];

    // ---- output layer: tanh(h3 @ Wout + bout), N padded to 16 ----
    {
      v8f acc = {0.f, 0.f, 0.f, 0.f, 0.f, 0.f, 0.f, 0.f};
#pragma unroll
      for (int kb = 0; kb < 8; ++kb)
        acc = __builtin_amdgcn_wmma_f32_16x16x4_f32(
            false, aH[kb], false, fWo[kb], (short)0, acc, false, false);
      if (n0 < 3) {
#pragma unroll
        for (int i = 0; i < 8; ++i) {
          float v = apply_act(acc[i] + cOut.bias, cOut);
          out[(size_t)(p0 + i + 8 * kh) * 3 + n0] = v;
        }
      }
    }
    __builtin_amdgcn_wave_barrier();
  }
}

extern "C" void kernel_launch(void* const* d_in, const int* in_sizes, int n_in,
                              void* d_out, int out_size, void* d_ws, size_t ws_size,
                              hipStream_t stream) {
  const float* inputs  = (const float*)d_in[0];
  const float* bias_in = (const float*)d_in[1];
  const float* W1 = (const float*)d_in[2];
  const float* b1 = (const float*)d_in[3];
  const int*   a1 = (const int*)  d_in[4];
  const float* W2 = (const float*)d_in[5];
  const float* b2 = (const float*)d_in[6];
  const int*   a2 = (const int*)  d_in[7];
  const float* W3 = (const float*)d_in[8];
  const float* b3 = (const float*)d_in[9];
  const int*   a3 = (const int*)  d_in[10];
  const float* Wo = (const float*)d_in[11];
  const float* bo = (const float*)d_in[12];
  float* out = (float*)d_out;

  dim3 grid(1024), block(256);   // 8192 waves, 8 tiles each (grid-stride)
  hipLaunchKernelGGL(cppn_kernel, grid, block, 0, stream,
                     inputs, bias_in, W1, b1, a1, W2, b2, a2, W3, b3, a3, Wo, bo, out);
}